// Proposal_20169166422189
// MI455X (gfx1250) — compile-verified
//
#include <hip/hip_runtime.h>
#include <hip/hip_bf16.h>
#include <stdint.h>

// ---------------- problem constants (fixed by reference setup_inputs) -------
#define BATCH   4
#define NA      9
#define HH      34
#define WW      60
#define NANCH   (NA * HH * WW)     // 18360 anchors per batch
#define PRE     3000               // PRE_NMS_TOP_N
#define PREP    3008               // padded to multiple of 32/16
#define POST    300                // POST_NMS_TOP_N
#define NWORDS  (PREP / 32)        // 94 mask words per row
#define IOU_TH  0.7f
#define IMG_W1  959.0f             // img_w - 1
#define IMG_H1  543.0f             // img_h - 1
#define SORT_N  32768              // next pow2 >= 18360

typedef __attribute__((ext_vector_type(2))) float v2f;
typedef __attribute__((ext_vector_type(8))) float v8f;
typedef __attribute__((ext_vector_type(4))) int   v4i;

#if __has_builtin(__builtin_amdgcn_wmma_f32_16x16x4_f32)
#define HAVE_WMMA_F32X4 1
#endif
#if __has_builtin(__builtin_amdgcn_global_load_async_to_lds_b128)
#define HAVE_ASYNC_LDS 1
typedef __attribute__((address_space(1))) v4i v4i_g;   // global int4
typedef __attribute__((address_space(3))) v4i v4i_l;   // LDS int4
#endif

static __device__ __forceinline__ void wait_async_lds() {
#if __has_builtin(__builtin_amdgcn_s_wait_asynccnt)
  __builtin_amdgcn_s_wait_asynccnt(0);
#else
  asm volatile("s_wait_asynccnt 0" ::: "memory");
#endif
}

// ---------------------------------------------------------------------------
// Kernel 1: decode all B*18360 boxes (elementwise; memory-bound, trivial)
// ---------------------------------------------------------------------------
__global__ void decode_kernel(const float* __restrict__ deltas,
                              float4* __restrict__ boxesDec) {
  int t = blockIdx.x * blockDim.x + threadIdx.x;
  if (t >= BATCH * NANCH) return;
  int b = t / NANCH, r = t % NANCH;
  int a = r / (HH * WW);
  int hw = r % (HH * WW);
  int h = hw / WW, w = hw % WW;

  const float sizes[3] = {64.f, 128.f, 256.f};
  const float rsq[3]   = {0.70710678118654752f, 1.0f, 1.41421356237309515f};
  float aw = sizes[a / 3] * rsq[a % 3];
  float ah = sizes[a / 3] / rsq[a % 3];

  const size_t HW = (size_t)HH * WW;
  size_t dbase = (((size_t)b * (4 * NA) + a * 4) * HH + h) * WW + w;
  float tx = deltas[dbase];
  float ty = deltas[dbase + HW];
  float tw = deltas[dbase + 2 * HW];
  float th = deltas[dbase + 3 * HW];

  float cx = (w + 0.5f) * 16.0f;
  float cy = (h + 0.5f) * 16.0f;
  float ctrx = cx + tx * aw;
  float ctry = cy + ty * ah;
  float w1 = expf(tw) * aw;
  float h1 = expf(th) * ah;
  float x1 = ctrx - 0.5f * w1;
  float y1 = ctry - 0.5f * h1;
  float x2 = x1 + w1;
  float y2 = y1 + h1;
  x1 = fminf(fmaxf(x1, 0.f), IMG_W1);
  x2 = fminf(fmaxf(x2, 0.f), IMG_W1);
  y1 = fminf(fmaxf(y1, 0.f), IMG_H1);
  y2 = fminf(fmaxf(y2, 0.f), IMG_H1);
  boxesDec[t] = make_float4(x1, y1, x2, y2);
}

// ---------------------------------------------------------------------------
// Kernel 2: per-batch full in-LDS bitonic sort (desc score, asc index ties).
// 32768 u64 keys = 256 KB LDS -> exploits CDNA5's 320 KB WGP LDS.
// One workgroup per batch; then gathers top-3008 boxes (pad = zero box).
// ---------------------------------------------------------------------------
__global__ __launch_bounds__(1024) void sort_kernel(
    const float* __restrict__ scores,
    const float4* __restrict__ boxesDec,
    float4* __restrict__ boxesK) {
  extern __shared__ unsigned long long s[];
  const int b = blockIdx.x;
  const unsigned tid = threadIdx.x;
  const unsigned n = SORT_N;

  for (unsigned i = tid; i < n; i += 1024) {
    unsigned long long key = 0ull;
    if (i < NANCH) {
      unsigned bits = __float_as_uint(scores[(size_t)b * NANCH + i]);
      bits = (bits & 0x80000000u) ? ~bits : (bits | 0x80000000u);
      key = ((unsigned long long)bits << 32) | (unsigned)(~i);  // ties: lower idx first
    }
    s[i] = key;
  }
  __syncthreads();

  for (unsigned k = 2; k <= n; k <<= 1) {
    for (unsigned j = k >> 1; j > 0; j >>= 1) {
      for (unsigned i = tid; i < n; i += 1024) {
        unsigned l = i ^ j;
        if (l > i) {
          unsigned long long x = s[i], y = s[l];
          bool sw = ((i & k) == 0) ? (x < y) : (x > y);  // descending sort
          if (sw) { s[i] = y; s[l] = x; }
        }
      }
      __syncthreads();
    }
  }

  for (unsigned t = tid; t < PREP; t += 1024) {
    float4 bx = make_float4(0.f, 0.f, 0.f, 0.f);
    if (t < PRE) {
      unsigned idx = ~(unsigned)(s[t] & 0xFFFFFFFFull);
      bx = boxesDec[(size_t)b * NANCH + idx];
    }
    boxesK[(size_t)b * PREP + t] = bx;
  }
}

// ---------------------------------------------------------------------------
// Kernel 3: NMS suppression bitmask. One wave per 16-row x 32-col tile.
// union = (area_i + area_j) - inter ; the rank-2 outer sum area_i+area_j is
// computed with V_WMMA_F32_16X16X4_F32 (A=[area,1|0,0], B=[1;area;0;0]).
// Box tiles staged to LDS with global_load_async_to_lds_b128; bits packed
// cross-lane via wave32 ballot (no atomics). IoU compare in multiply form:
// inter/max(u,eps) > t  <=>  inter > t*max(u,eps)   (divisor > 0).
// ---------------------------------------------------------------------------
__global__ __launch_bounds__(32) void nms_mask_kernel(
    const float4* __restrict__ boxesK, unsigned* __restrict__ mask) {
  const int lane = threadIdx.x;
  const int jw = blockIdx.x;             // col word 0..93   (cols j0..j0+31)
  const int it = blockIdx.y;             // row tile 0..187  (rows i0..i0+15)
  const int b  = blockIdx.z;
  const int i0 = it * 16;
  const int j0 = jw * 32;
  const size_t rowbase = (size_t)b * PREP + i0;

  if (j0 + 31 <= i0) {                   // strictly lower-triangular: no j>i bits
    if (lane < 16) mask[(rowbase + lane) * NWORDS + jw] = 0u;
    return;
  }

  __shared__ float4 rbS[16];
  __shared__ float4 cbS[32];
  const float4* rowsrc = boxesK + (size_t)b * PREP + i0;
  const float4* colsrc = boxesK + (size_t)b * PREP + j0;

#ifdef HAVE_ASYNC_LDS
  if (lane < 16)
    __builtin_amdgcn_global_load_async_to_lds_b128(
        (v4i_g*)(uintptr_t)(rowsrc + lane),
        (v4i_l*)(unsigned int)(uintptr_t)(rbS + lane), 0, 0);
  __builtin_amdgcn_global_load_async_to_lds_b128(
      (v4i_g*)(uintptr_t)(colsrc + lane),
      (v4i_l*)(unsigned int)(uintptr_t)(cbS + lane), 0, 0);
  wait_async_lds();
  __syncthreads();
#else
  if (lane < 16) rbS[lane] = rowsrc[lane];
  cbS[lane] = colsrc[lane];
  __syncthreads();
#endif

  const bool lo = lane < 16;
  const int nIdx = lane & 15;
  const int mhalf = (lane >> 4) << 3;    // 0 or 8 (C/D layout M offset)

  float4 rbl = rbS[nIdx];
  float areaR = (rbl.z - rbl.x) * (rbl.w - rbl.y);
  float4 cbA = cbS[nIdx];
  float4 cbB = cbS[16 + nIdx];
  float aC0 = (cbA.z - cbA.x) * (cbA.w - cbA.y);
  float aC1 = (cbB.z - cbB.x) * (cbB.w - cbB.y);

#ifdef HAVE_WMMA_F32X4
  v2f av;  av.x = lo ? areaR : 0.f; av.y = lo ? 1.f : 0.f;
  v2f bv0; bv0.x = lo ? 1.f : 0.f;  bv0.y = lo ? aC0 : 0.f;
  v2f bv1; bv1.x = lo ? 1.f : 0.f;  bv1.y = lo ? aC1 : 0.f;
  v8f cz = {};
  // S[v][lane] = area_row(M) + area_col(N), M = v + mhalf, N = lane&15
  v8f S0 = __builtin_amdgcn_wmma_f32_16x16x4_f32(false, av, false, bv0, (short)0, cz, false, false);
  v8f S1 = __builtin_amdgcn_wmma_f32_16x16x4_f32(false, av, false, bv1, (short)0, cz, false, false);
#endif

#pragma unroll
  for (int v = 0; v < 8; ++v) {
    float4 rb = rbS[v + mhalf];
    int gi = i0 + v + mhalf;
#ifndef HAVE_WMMA_F32X4
    float arR = (rb.z - rb.x) * (rb.w - rb.y);
#endif
    // cols j0 + 0..15
    float iw0 = fmaxf(fminf(rb.z, cbA.z) - fmaxf(rb.x, cbA.x), 0.f);
    float ih0 = fmaxf(fminf(rb.w, cbA.w) - fmaxf(rb.y, cbA.y), 0.f);
    float inter0 = iw0 * ih0;
#ifdef HAVE_WMMA_F32X4
    float uni0 = S0[v] - inter0;
#else
    float uni0 = arR + aC0 - inter0;
#endif
    bool p0 = (inter0 > IOU_TH * fmaxf(uni0, 1e-8f)) && ((j0 + nIdx) > gi);

    // cols j0 + 16..31
    float iw1 = fmaxf(fminf(rb.z, cbB.z) - fmaxf(rb.x, cbB.x), 0.f);
    float ih1 = fmaxf(fminf(rb.w, cbB.w) - fmaxf(rb.y, cbB.y), 0.f);
    float inter1 = iw1 * ih1;
#ifdef HAVE_WMMA_F32X4
    float uni1 = S1[v] - inter1;
#else
    float uni1 = arR + aC1 - inter1;
#endif
    bool p1 = (inter1 > IOU_TH * fmaxf(uni1, 1e-8f)) && ((j0 + 16 + nIdx) > gi);

    unsigned m0 = (unsigned)__ballot(p0);  // bits 0-15: row v | bits 16-31: row v+8
    unsigned m1 = (unsigned)__ballot(p1);
    if (lane == 0)
      mask[(rowbase + v) * NWORDS + jw] = (m0 & 0xFFFFu) | ((m1 & 0xFFFFu) << 16);
    if (lane == 16)
      mask[(rowbase + v + 8) * NWORDS + jw] = (m0 >> 16) | (m1 & 0xFFFF0000u);
  }
}

// ---------------------------------------------------------------------------
// Kernel 4: sequential greedy reduce (exactly the reference scan semantics).
// Single wave32 per batch; 94-word remv bitset in LDS.
// ---------------------------------------------------------------------------
__global__ __launch_bounds__(32) void nms_reduce_kernel(
    const unsigned* __restrict__ mask, unsigned* __restrict__ keep) {
  const int b = blockIdx.x;
  const int lane = threadIdx.x;
  __shared__ unsigned remv[NWORDS];
  for (int w = lane; w < NWORDS; w += 32) remv[w] = 0u;
  __syncthreads();
  for (int i = 0; i < PRE; ++i) {
    unsigned cur = remv[i >> 5];                 // wave-uniform
    bool kept = ((cur >> (i & 31)) & 1u) == 0u;
    if (kept) {
      const unsigned* mrow = &mask[((size_t)b * PREP + i) * NWORDS];
      for (int w = lane; w < NWORDS; w += 32) remv[w] |= mrow[w];
    }
    if (lane == 0) keep[(size_t)b * PREP + i] = kept ? 1u : 0u;
    __syncthreads();
  }
}

// ---------------------------------------------------------------------------
// Kernel 5: emit first 300 kept boxes in order (ballot prefix compaction),
// zero-fill the rest. Single wave32 per batch.
// ---------------------------------------------------------------------------
__global__ __launch_bounds__(32) void gather_out_kernel(
    const float4* __restrict__ boxesK, const unsigned* __restrict__ keep,
    float* __restrict__ out) {
  const int b = blockIdx.x;
  const int lane = threadIdx.x;
  float* ob = out + (size_t)b * POST * 4;
  for (int t = lane; t < POST * 4; t += 32) ob[t] = 0.f;
  __syncthreads();
  int base = 0;
  for (int i0 = 0; i0 < PRE && base < POST; i0 += 32) {
    int i = i0 + lane;
    bool p = (i < PRE) && (keep[(size_t)b * PREP + i] != 0u);
    unsigned m = (unsigned)__ballot(p);
    int rank = base + __popc(m & ((1u << lane) - 1u));
    if (p && rank < POST) {
      float4 bx = boxesK[(size_t)b * PREP + i];
      ob[rank * 4 + 0] = bx.x;
      ob[rank * 4 + 1] = bx.y;
      ob[rank * 4 + 2] = bx.z;
      ob[rank * 4 + 3] = bx.w;
    }
    base += __popc(m);
  }
}

// ---------------------------------------------------------------------------
extern "C" void kernel_launch(void* const* d_in, const int* in_sizes, int n_in,
                              void* d_out, int out_size, void* d_ws, size_t ws_size,
                              hipStream_t stream) {
  const float* scores = (const float*)d_in[0];   // (4,9,34,60)
  const float* deltas = (const float*)d_in[1];   // (4,36,34,60)
  (void)in_sizes; (void)n_in; (void)out_size; (void)ws_size;
  float* out = (float*)d_out;                    // (4,300,4)

  char* ws = (char*)d_ws;
  size_t off = 0;
  float4* boxesDec = (float4*)(ws + off); off += (size_t)BATCH * NANCH * sizeof(float4);
  float4* boxesK   = (float4*)(ws + off); off += (size_t)BATCH * PREP * sizeof(float4);
  unsigned* mask   = (unsigned*)(ws + off); off += (size_t)BATCH * PREP * NWORDS * sizeof(unsigned);
  unsigned* keep   = (unsigned*)(ws + off); off += (size_t)BATCH * PREP * sizeof(unsigned);

  // 1) decode all boxes
  {
    int total = BATCH * NANCH;
    decode_kernel<<<(total + 255) / 256, 256, 0, stream>>>(deltas, boxesDec);
  }
  // 2) per-batch in-LDS bitonic top-3000 (256 KB dynamic LDS per workgroup)
  sort_kernel<<<BATCH, 1024, SORT_N * sizeof(unsigned long long), stream>>>(
      scores, boxesDec, boxesK);
  // 3) suppression bitmask (WMMA outer-sum + ballot packing)
  {
    dim3 grid(NWORDS, PREP / 16, BATCH);
    nms_mask_kernel<<<grid, 32, 0, stream>>>(boxesK, mask);
  }
  // 4) sequential greedy reduce
  nms_reduce_kernel<<<BATCH, 32, 0, stream>>>(mask, keep);
  // 5) compact first 300 kept boxes
  gather_out_kernel<<<BATCH, 32, 0, stream>>>(boxesK, keep, out);
}